// InvariantPointMessagePassing_20779051778382
// MI455X (gfx1250) — compile-verified
//
#include <hip/hip_runtime.h>

// ---------------- problem constants (from reference setup_inputs) -----------
constexpr int BB = 8;      // batch
constexpr int NN = 512;    // nodes
constexpr int KK = 32;     // neighbors
constexpr int DD = 128;    // node feature dim
constexpr int HH = 128;    // hidden dim
constexpr int PP = 8;      // points
constexpr int MSG = 456;   // 3*D + 9*P
constexpr int MS1 = 460;   // padded LDS stride for msg rows (460 % 64 == 12 -> conflict-free b64 A loads)
constexpr int MS2 = 132;   // padded LDS stride for m1/m2/m3 rows
constexpr int GEO = 68;    // per-node geometry record: R[9] t[3] pl[24] pg[24] plnorm[8]

#define EPS   1e-8f
#define LNEPS 1e-5f
#define PSCALE 10.0f

typedef float v2f __attribute__((ext_vector_type(2)));
typedef float v8f __attribute__((ext_vector_type(8)));
typedef unsigned int v4u __attribute__((ext_vector_type(4)));
typedef int v4i __attribute__((ext_vector_type(4)));
typedef int v8i __attribute__((ext_vector_type(8)));

#define HAVE_WMMA_F32 __has_builtin(__builtin_amdgcn_wmma_f32_16x16x4_f32)
#define HAVE_TDM      __has_builtin(__builtin_amdgcn_tensor_load_to_lds)
#define HAVE_TENSWAIT __has_builtin(__builtin_amdgcn_s_wait_tensorcnt)

// ---------------- TDM: 1-D f32 copy global -> LDS (D# per ISA 8.3/8.4) ------
// 6-arg builtin (clang-23 / therock form): (g0, g1, g2, g3, g_extra, cpol)
#if HAVE_TDM
__device__ inline void tdm_load_f32_1d(const float* gsrc, unsigned lds_byte_off, unsigned n) {
  unsigned long long ga = (unsigned long long)(const void*)gsrc;
  v4u g0;
  g0[0] = 1u;                                              // count=1, user mode, no gather
  g0[1] = lds_byte_off;                                    // lds_addr (bytes)
  g0[2] = (unsigned)ga;                                    // global_addr[31:0]
  g0[3] = (unsigned)((ga >> 32) & 0x1FFFFFFull) | (2u << 30); // addr[56:32] | type=2
  v8i g1;
  g1[0] = (int)(2u << 16);                                 // data_size=2 (4 bytes)
  g1[1] = (int)((n & 0xFFFFu) << 16);                      // tensor_dim0 lo16 (bits 63:48)
  g1[2] = (int)(((n >> 16) & 0xFFFFu) | (1u << 16));       // tensor_dim0 hi | tensor_dim1=1 lo
  g1[3] = (int)((n & 0xFFFFu) << 16);                      // tensor_dim1 hi=0 | tile_dim0=n
  g1[4] = 1;                                               // tile_dim1=1, tile_dim2=0
  g1[5] = (int)n;                                          // tensor_dim0_stride lo32
  g1[6] = 0;
  g1[7] = 0;
  v4i z4 = {0, 0, 0, 0};
  v8i z8 = {0, 0, 0, 0, 0, 0, 0, 0};
  __builtin_amdgcn_tensor_load_to_lds(g0, g1, z4, z4, z8, 0);
}
#endif

// ---------------- kernel 0: h_E passthrough (tuple output #2) ---------------
__global__ void copy_hE_kernel(const float4* __restrict__ src, float4* __restrict__ dst, int n4) {
  int i = blockIdx.x * blockDim.x + threadIdx.x;
  if (i < n4) dst[i] = src[i];
}

// ---------------- kernel 1: backbone frames + points (1 wave / node) --------
__global__ __launch_bounds__(32) void frames_kernel(
    const float* __restrict__ hV, const float* __restrict__ X,
    const float* __restrict__ Wp, const float* __restrict__ bp,
    float* __restrict__ geo) {
  __shared__ float s_pl[24];
  const int bn = blockIdx.x;
  const int lane = threadIdx.x;
  const float* Xr = X + (size_t)bn * 9; // [atom][xyz], atoms: N, CA, C
  float nx = Xr[0], ny = Xr[1], nz = Xr[2];
  float ax = Xr[3], ay = Xr[4], az = Xr[5];
  float cx = Xr[6], cy = Xr[7], cz = Xr[8];
  float v1x = cx - ax, v1y = cy - ay, v1z = cz - az;
  float v2x = nx - ax, v2y = ny - ay, v2z = nz - az;
  float in1 = rsqrtf(v1x * v1x + v1y * v1y + v1z * v1z + EPS);
  float e1x = v1x * in1, e1y = v1y * in1, e1z = v1z * in1;
  float d = e1x * v2x + e1y * v2y + e1z * v2z;
  float u2x = v2x - e1x * d, u2y = v2y - e1y * d, u2z = v2z - e1z * d;
  float in2 = rsqrtf(u2x * u2x + u2y * u2y + u2z * u2z + EPS);
  float e2x = u2x * in2, e2y = u2y * in2, e2z = u2z * in2;
  float e3x = e1y * e2z - e1z * e2y;
  float e3y = e1z * e2x - e1x * e2z;
  float e3z = e1x * e2y - e1y * e2x;
  float t0 = ax / PSCALE, t1 = ay / PSCALE, t2 = az / PSCALE;
  float* rec = geo + (size_t)bn * GEO;
  if (lane == 0) { // R row-major R[i][j], columns = e1,e2,e3
    rec[0] = e1x; rec[1] = e2x; rec[2] = e3x;
    rec[3] = e1y; rec[4] = e2y; rec[5] = e3y;
    rec[6] = e1z; rec[7] = e2z; rec[8] = e3z;
    rec[9] = t0; rec[10] = t1; rec[11] = t2;
  }
  // p_local = hV @ Wp + bp  (24 outputs, one per lane)
  if (lane < 24) {
    const float* hv = hV + (size_t)bn * DD;
    float acc = bp[lane];
    for (int k = 0; k < DD; ++k) acc = fmaf(hv[k], Wp[k * 24 + lane], acc);
    s_pl[lane] = acc;
    rec[12 + lane] = acc;
  }
  __syncthreads();
  if (lane < 24) { // p_global[i] = sum_j R[i][j]*pl[j] + t[i]
    int p = lane / 3, i = lane % 3;
    float p0 = s_pl[p * 3 + 0], p1 = s_pl[p * 3 + 1], p2 = s_pl[p * 3 + 2];
    float r0 = (i == 0) ? e1x : ((i == 1) ? e1y : e1z);
    float r1 = (i == 0) ? e2x : ((i == 1) ? e2y : e2z);
    float r2 = (i == 0) ? e3x : ((i == 1) ? e3y : e3z);
    float ti = (i == 0) ? t0 : ((i == 1) ? t1 : t2);
    rec[36 + lane] = r0 * p0 + r1 * p1 + r2 * p2 + ti;
  }
  if (lane < PP) {
    float a = s_pl[lane * 3], b = s_pl[lane * 3 + 1], c = s_pl[lane * 3 + 2];
    rec[60 + lane] = sqrtf(a * a + b * b + c * c + EPS);
  }
}

// ---------------- WMMA helper: [32 x KD] (LDS) @ [KD x 128] (global) --------
// Wave w owns columns [32w, 32w+32); 2 row-tiles x 2 col-tiles of 16x16.
// A 16x4 f32 layout (ISA 7.12.2): VGPR0 = K {0|2}, VGPR1 = K {1|3}; lanes 16-31 carry K+2.
template <int KD>
__device__ inline void wmma_layer(const float* __restrict__ A_lds, int astride,
                                  const float* __restrict__ W,
                                  const float* __restrict__ bias_lds,
                                  float* __restrict__ out_lds, int ostride,
                                  int wid, int lane, bool relu,
                                  const float* __restrict__ rowmask) {
  const int n0 = wid * 32;
  const int l15 = lane & 15;
#if HAVE_WMMA_F32
  const int sel2 = (lane >> 4) << 1; // 0 or 2
  v8f a00 = {0.f,0.f,0.f,0.f,0.f,0.f,0.f,0.f};
  v8f a01 = a00, a10 = a00, a11 = a00;
  for (int kk = 0; kk < KD; kk += 4) {
    int kc = kk + sel2;
    v2f va0 = *(const v2f*)(A_lds + l15 * astride + kc);
    v2f va1 = *(const v2f*)(A_lds + (l15 + 16) * astride + kc);
    const float* wp = W + (size_t)kc * HH + n0 + l15;
    v2f vb0 = { wp[0], wp[HH] };
    v2f vb1 = { wp[16], wp[HH + 16] };
    a00 = __builtin_amdgcn_wmma_f32_16x16x4_f32(false, va0, false, vb0, (short)0, a00, false, false);
    a01 = __builtin_amdgcn_wmma_f32_16x16x4_f32(false, va0, false, vb1, (short)0, a01, false, false);
    a10 = __builtin_amdgcn_wmma_f32_16x16x4_f32(false, va1, false, vb0, (short)0, a10, false, false);
    a11 = __builtin_amdgcn_wmma_f32_16x16x4_f32(false, va1, false, vb1, (short)0, a11, false, false);
  }
  __syncthreads(); // all waves done reading A before out (may alias A region)
  float b0 = bias_lds[n0 + l15];
  float b1 = bias_lds[n0 + 16 + l15];
  const int radd = (lane >> 4) << 3; // C layout: lanes 16-31 carry rows M+8
#pragma unroll
  for (int vg = 0; vg < 8; ++vg) {
    int r0 = vg + radd, r1 = r0 + 16;
    float v00 = a00[vg] + b0, v01 = a01[vg] + b1;
    float v10 = a10[vg] + b0, v11 = a11[vg] + b1;
    if (relu) {
      v00 = fmaxf(v00, 0.f); v01 = fmaxf(v01, 0.f);
      v10 = fmaxf(v10, 0.f); v11 = fmaxf(v11, 0.f);
    }
    if (rowmask) {
      float m0 = rowmask[r0], m1 = rowmask[r1];
      v00 *= m0; v01 *= m0; v10 *= m1; v11 *= m1;
    }
    out_lds[r0 * ostride + n0 + l15]      = v00;
    out_lds[r0 * ostride + n0 + 16 + l15] = v01;
    out_lds[r1 * ostride + n0 + l15]      = v10;
    out_lds[r1 * ostride + n0 + 16 + l15] = v11;
  }
  __syncthreads();
#else
  // VALU fallback: thread (wid,lane) -> rows wid*8..+7, cols lane + 32j
  float vals[8][4];
#pragma unroll
  for (int i = 0; i < 8; ++i)
#pragma unroll
    for (int j = 0; j < 4; ++j) {
      int r = wid * 8 + i, c = lane + 32 * j;
      float acc = bias_lds[c];
      for (int k = 0; k < KD; ++k) acc = fmaf(A_lds[r * astride + k], W[(size_t)k * HH + c], acc);
      if (relu) acc = fmaxf(acc, 0.f);
      if (rowmask) acc *= rowmask[r];
      vals[i][j] = acc;
    }
  __syncthreads();
#pragma unroll
  for (int i = 0; i < 8; ++i)
#pragma unroll
    for (int j = 0; j < 4; ++j)
      out_lds[(wid * 8 + i) * ostride + lane + 32 * j] = vals[i][j];
  __syncthreads();
#endif
  (void)n0; (void)l15;
}

// ---------------- kernel 2: message MLP (1 workgroup / node) ----------------
__global__ __launch_bounds__(128) void msg_mlp_kernel(
    const float* __restrict__ hV, const float* __restrict__ hE,
    const int* __restrict__ Eidx, const float* __restrict__ maskA,
    const float* __restrict__ Wm1, const float* __restrict__ bm1,
    const float* __restrict__ Wm2, const float* __restrict__ bm2,
    const float* __restrict__ Wm3, const float* __restrict__ bm3,
    const float* __restrict__ geo, float* __restrict__ node_m) {
  // single LDS block (assumed at LDS offset 0 for the TDM descriptor)
  __shared__ float sm[KK * MS1 + GEO + 3 * HH]; // 14720 + 68 + 384 floats = 60688 B
  float* s_msg = sm;
  float* s_ctr = sm + KK * MS1;
  float* s_bias = s_ctr + GEO;
  const unsigned bias_lds_off = (unsigned)((KK * MS1 + GEO) * 4);

  const int bn = blockIdx.x;
  const int b = bn / NN;
  const int tid = threadIdx.x;
  const int wid = tid >> 5, lane = tid & 31;

  // stage biases into LDS via the Tensor Data Mover (wave 0 issues; TENSORcnt)
  if (wid == 0) {
#if HAVE_TDM
    tdm_load_f32_1d(bm1, bias_lds_off, HH);
    tdm_load_f32_1d(bm2, bias_lds_off + HH * 4, HH);
    tdm_load_f32_1d(bm3, bias_lds_off + 2 * HH * 4, HH);
#if HAVE_TENSWAIT
    __builtin_amdgcn_s_wait_tensorcnt(0);
#endif
#else
    for (int c = lane; c < HH; c += 32) {
      s_bias[c] = bm1[c];
      s_bias[HH + c] = bm2[c];
      s_bias[2 * HH + c] = bm3[c];
    }
#endif
  }
  if (tid < GEO) s_ctr[tid] = geo[(size_t)bn * GEO + tid];
  __syncthreads();

  // ---- build msg_in[32][456] : [hV(128) | hE(128) | nbr_hV(128) | pl(24) |
  //                              plnorm(8) | nbr_pl(24) | nplnorm(8) | npgnorm(8)]
  const float* hvc = hV + (size_t)bn * DD;
  for (int k = wid; k < KK; k += 4) {
    float* row = s_msg + k * MS1;
    int j = Eidx[(size_t)bn * KK + k];
    const float* hvn = hV + ((size_t)b * NN + j) * DD;
    const float* hec = hE + ((size_t)bn * KK + k) * DD;
    for (int c = lane; c < DD; c += 32) {
      row[c] = hvc[c];
      row[DD + c] = hec[c];
      row[2 * DD + c] = hvn[c];
    }
    if (lane < 24) row[384 + lane] = s_ctr[12 + lane];           // pl_flat
    else           row[408 + (lane - 24)] = s_ctr[60 + (lane - 24)]; // pl_norm
    if (lane < PP) {
      const float* nrec = geo + ((size_t)b * NN + j) * GEO;
      float gx = nrec[36 + lane * 3 + 0];
      float gy = nrec[36 + lane * 3 + 1];
      float gz = nrec[36 + lane * 3 + 2];
      float dx = s_ctr[36 + lane * 3 + 0] - gx;
      float dy = s_ctr[36 + lane * 3 + 1] - gy;
      float dz = s_ctr[36 + lane * 3 + 2] - gz;
      row[448 + lane] = sqrtf(dx * dx + dy * dy + dz * dz + EPS); // npg_norm
      float qx = gx - s_ctr[9], qy = gy - s_ctr[10], qz = gz - s_ctr[11];
      // nbr_pl[j] = sum_i R[i][j] * q[i]   (R row-major in s_ctr[0..8])
      float p0 = s_ctr[0] * qx + s_ctr[3] * qy + s_ctr[6] * qz;
      float p1 = s_ctr[1] * qx + s_ctr[4] * qy + s_ctr[7] * qz;
      float p2 = s_ctr[2] * qx + s_ctr[5] * qy + s_ctr[8] * qz;
      row[416 + lane * 3 + 0] = p0;
      row[416 + lane * 3 + 1] = p1;
      row[416 + lane * 3 + 2] = p2;
      row[440 + lane] = sqrtf(p0 * p0 + p1 * p1 + p2 * p2 + EPS); // npl_norm
    }
  }
  __syncthreads();

  // ---- 3-layer MLP with fp32 WMMA (16x16x4). LDS region reuse:
  //   m1 -> float offset 0 (msg rows fully consumed first), m2 -> 32*MS2, m3 -> 0
  wmma_layer<MSG>(s_msg, MS1, Wm1, s_bias, sm, MS2, wid, lane, true, nullptr);
  wmma_layer<HH>(sm, MS2, Wm2, s_bias + HH, sm + KK * MS2, MS2, wid, lane, true, nullptr);
  wmma_layer<HH>(sm + KK * MS2, MS2, Wm3, s_bias + 2 * HH, sm, MS2, wid, lane, false,
                 maskA + (size_t)bn * KK);

  // ---- masked mean over K (mask already applied per-row in layer 3)
  float s = 0.f;
#pragma unroll
  for (int r = 0; r < KK; ++r) s += sm[r * MS2 + tid];
  node_m[(size_t)bn * HH + tid] = s * (1.0f / (float)KK);
}

// ---------------- kernel 3: residual + LN + FFN + LN + mask -----------------
__device__ inline float blk_sum128(float v, float* red, int t) {
  red[t] = v;
  __syncthreads();
  for (int s = 64; s > 0; s >>= 1) {
    if (t < s) red[t] += red[t + s];
    __syncthreads();
  }
  float r = red[0];
  __syncthreads();
  return r;
}

__global__ __launch_bounds__(128) void node_update_kernel(
    const float* __restrict__ hV, const float* __restrict__ node_m,
    const float* __restrict__ maskV,
    const float* __restrict__ g1, const float* __restrict__ be1,
    const float* __restrict__ Wd1, const float* __restrict__ bd1,
    const float* __restrict__ Wd2, const float* __restrict__ bd2,
    const float* __restrict__ g2, const float* __restrict__ be2,
    float* __restrict__ out) {
  __shared__ float sh[HH], dh[4 * HH], red[HH];
  const int bn = blockIdx.x, t = threadIdx.x;
  float x = hV[(size_t)bn * HH + t] + node_m[(size_t)bn * HH + t];
  float mu = blk_sum128(x, red, t) * (1.0f / HH);
  float c = x - mu;
  float var = blk_sum128(c * c, red, t) * (1.0f / HH);
  float h = c * rsqrtf(var + LNEPS) * g1[t] + be1[t];
  sh[t] = h;
  __syncthreads();
  for (int u = t; u < 4 * HH; u += HH) {
    float a = bd1[u];
    for (int k = 0; k < HH; ++k) a = fmaf(sh[k], Wd1[(size_t)k * (4 * HH) + u], a);
    dh[u] = fmaxf(a, 0.f);
  }
  __syncthreads();
  float a = bd2[t];
  for (int k = 0; k < 4 * HH; ++k) a = fmaf(dh[k], Wd2[(size_t)k * HH + t], a);
  float x2 = h + a;
  float mu2 = blk_sum128(x2, red, t) * (1.0f / HH);
  float c2 = x2 - mu2;
  float var2 = blk_sum128(c2 * c2, red, t) * (1.0f / HH);
  float o = c2 * rsqrtf(var2 + LNEPS) * g2[t] + be2[t];
  out[(size_t)bn * HH + t] = o * maskV[bn];
}

// ---------------- host launch ----------------------------------------------
extern "C" void kernel_launch(void* const* d_in, const int* in_sizes, int n_in,
                              void* d_out, int out_size, void* d_ws, size_t ws_size,
                              hipStream_t stream) {
  const float* hV    = (const float*)d_in[0];
  const float* hE    = (const float*)d_in[1];
  const float* X     = (const float*)d_in[2];
  const float* maskV = (const float*)d_in[3];
  const float* maskA = (const float*)d_in[4];
  const float* Wp    = (const float*)d_in[5];
  const float* bp    = (const float*)d_in[6];
  const float* Wm1   = (const float*)d_in[7];
  const float* bm1   = (const float*)d_in[8];
  const float* Wm2   = (const float*)d_in[9];
  const float* bm2   = (const float*)d_in[10];
  const float* Wm3   = (const float*)d_in[11];
  const float* bm3   = (const float*)d_in[12];
  const float* g1    = (const float*)d_in[13];
  const float* be1   = (const float*)d_in[14];
  const float* Wd1   = (const float*)d_in[15];
  const float* bd1   = (const float*)d_in[16];
  const float* Wd2   = (const float*)d_in[17];
  const float* bd2   = (const float*)d_in[18];
  const float* g2    = (const float*)d_in[19];
  const float* be2   = (const float*)d_in[20];
  const int*   Eidx  = (const int*)d_in[21];

  float* out  = (float*)d_out;                    // tuple: [h (B*N*H)] [h_E (B*N*K*D)]
  float* outE = out + (size_t)BB * NN * HH;
  float* geo    = (float*)d_ws;                   // B*N*GEO floats
  float* node_m = geo + (size_t)BB * NN * GEO;    // B*N*H floats

  const int nodes = BB * NN;
  const int n4 = BB * NN * KK * DD / 4;
  copy_hE_kernel<<<(n4 + 255) / 256, 256, 0, stream>>>((const float4*)hE, (float4*)outE, n4);
  frames_kernel<<<nodes, 32, 0, stream>>>(hV, X, Wp, bp, geo);
  msg_mlp_kernel<<<nodes, 128, 0, stream>>>(hV, hE, Eidx, maskA,
                                            Wm1, bm1, Wm2, bm2, Wm3, bm3,
                                            geo, node_m);
  node_update_kernel<<<nodes, 128, 0, stream>>>(hV, node_m, maskV,
                                                g1, be1, Wd1, bd1, Wd2, bd2, g2, be2, out);
}